// Seg_Sele_52544629899867
// MI455X (gfx1250) — compile-verified
//
#include <hip/hip_runtime.h>
#include <hip/hip_bf16.h>
#include <math.h>

typedef __attribute__((ext_vector_type(2))) float v2f;
typedef __attribute__((ext_vector_type(8))) float v8f;

#define D512 512
#define NB 64          // batch
#define NS 8           // segments
#define NF 4           // frames per segment
#define NT 64          // T
// visual strides (floats): b: 32*64*512 = 1048576, nf: 64*512 = 32768, t: 512
#define VIS_B 1048576
#define VIS_NF 32768
#define TOPSEG_ELEMS (2*NB*NF*NT*D512)   // 16777216
#define SEGFEA_ELEMS (NB*NS*D512)        // 262144

// ---------------- Kernel 1: seg_fea = mean over F of visual[:, :, 0, :] ----------------
__global__ __launch_bounds__(256) void segfea_kernel(const float* __restrict__ visual,
                                                     float* __restrict__ seg_fea) {
    int i = blockIdx.x * 256 + threadIdx.x;      // 0 .. 262143
    int d = i & 511;
    int s = (i >> 9) & 7;
    int b = i >> 12;
    const float* base = visual + (size_t)b * VIS_B + (size_t)(s * NF) * VIS_NF + d;
    float sum = base[0] + base[VIS_NF] + base[2 * VIS_NF] + base[3 * VIS_NF];
    seg_fea[i] = sum * 0.25f;
}

// ---------------- Kernel 2: C[M,512] = A[M,512] @ W^T[512,512] + bias, via WMMA f32 ----------------
// One wave per 16x16 output tile. K-loop in steps of 4 with V_WMMA_F32_16X16X4_F32.
__global__ __launch_bounds__(256) void gemm_wmma_kernel(const float* __restrict__ A,
                                                        const float* __restrict__ W,
                                                        const float* __restrict__ bias,
                                                        float* __restrict__ C,
                                                        int M) {
    const int tilesN = D512 / 16;                       // 32
    int wave = blockIdx.x * (256 / 32) + (threadIdx.x >> 5);
    int tm = (wave / tilesN) * 16;
    int tn = (wave % tilesN) * 16;
    if (tm >= M) return;                                // uniform per wave; active waves keep EXEC all-1s

    int lane = threadIdx.x & 31;
    int mn = lane & 15;            // A-frag row (M) / B-frag & C-frag column (N)
    int kh = (lane >> 4) << 1;     // which K pair within the 4-wide slab

    const float* Arow = A + (size_t)(tm + mn) * D512 + kh;   // A[tm+mn, k0+kh .. +1]
    const float* Wrow = W + (size_t)(tn + mn) * D512 + kh;   // B[k,n] = W[n,k] -> W[tn+mn, k0+kh .. +1]

    v8f acc = {};
#pragma unroll 4
    for (int k0 = 0; k0 < D512; k0 += 4) {
        v2f a = *(const v2f*)(Arow + k0);
        v2f b = *(const v2f*)(Wrow + k0);
        // 8 args: (neg_a, A, neg_b, B, c_mod, C, reuse_a, reuse_b)
        acc = __builtin_amdgcn_wmma_f32_16x16x4_f32(false, a, false, b,
                                                    (short)0, acc, false, false);
    }

    float bv = bias[tn + mn];
    int mbase = (lane >> 4) << 3;  // lanes 0-15: rows v; lanes 16-31: rows v+8
#pragma unroll
    for (int v = 0; v < 8; ++v) {
        C[(size_t)(tm + mbase + v) * D512 + tn + mn] = acc[v] + bv;
    }
}

// ---------------- Kernel 3: scores + gumbel argmax ----------------
// scores[b,s] = (Q[b] . K[b*8+s]) / sqrt(512); index[k,b] = argmax_s(scores[b,s] + g[k,b,s])
// (softmax and the per-row logsumexp shift cannot change the argmax, so they are skipped)
__global__ __launch_bounds__(512) void score_kernel(const float* __restrict__ Q,
                                                    const float* __restrict__ K,
                                                    const float* __restrict__ y_gumbel,
                                                    int* __restrict__ index) {
    __shared__ float scores[NB][NS];
    int b = threadIdx.x >> 3;
    int s = threadIdx.x & 7;
    const float* q = Q + (size_t)b * D512;
    const float* k = K + (size_t)(b * NS + s) * D512;
    float acc = 0.0f;
#pragma unroll 8
    for (int d = 0; d < D512; ++d) acc = fmaf(q[d], k[d], acc);
    scores[b][s] = acc * 0.044194173824159216f;   // 1/sqrt(512)
    __syncthreads();

    if (threadIdx.x < 2 * NB) {
        int kk = threadIdx.x >> 6;   // top_k slot
        int bb = threadIdx.x & 63;
        int best = 0;
        float bestv = -INFINITY;
#pragma unroll
        for (int s2 = 0; s2 < NS; ++s2) {
            float y = y_gumbel[((size_t)kk * NB + bb) * NS + s2];
            float g = -logf(-logf(y + 1e-25f));
            float v = scores[bb][s2] + g;
            if (v > bestv) { bestv = v; best = s2; }   // '>' keeps first max, matching jnp.argmax
        }
        index[kk * NB + bb] = best;
    }
}

// ---------------- Kernel 4: bandwidth-bound gather copy (float4 / B128 path) ----------------
// top_seg[b,k,f,t,d] = visual[b, index[k,b]*4 + f, t, d]
__global__ __launch_bounds__(256) void gather_kernel(const float4* __restrict__ visual,
                                                     const int* __restrict__ index,
                                                     float4* __restrict__ out) {
    long i = (long)blockIdx.x * 256 + threadIdx.x;    // 0 .. 4194303 float4s
    int d4 = (int)(i & 8191);                          // t*128 + d/4  (8192 float4 per (b,k,f))
    int f  = (int)(i >> 13) & 3;
    int kk = (int)(i >> 15) & 1;
    int b  = (int)(i >> 16);
    int s = index[kk * NB + b];                        // broadcast, L2-cached
    const float4* src = visual + (size_t)b * (VIS_B / 4)
                               + (size_t)(s * NF + f) * (VIS_NF / 4) + d4;
    out[i] = *src;
}

extern "C" void kernel_launch(void* const* d_in, const int* in_sizes, int n_in,
                              void* d_out, int out_size, void* d_ws, size_t ws_size,
                              hipStream_t stream) {
    const float* query    = (const float*)d_in[0];   // (64, 512)
    const float* visual   = (const float*)d_in[1];   // (64, 32, 64, 512)
    const float* Wq       = (const float*)d_in[2];   // (512, 512)
    const float* bq       = (const float*)d_in[3];   // (512,)
    const float* Ws       = (const float*)d_in[4];   // (512, 512)
    const float* bs       = (const float*)d_in[5];   // (512,)
    const float* y_gumbel = (const float*)d_in[6];   // (2, 64, 8)

    float* out      = (float*)d_out;
    float* top_seg  = out;                           // 16777216 floats
    float* seg_fea  = out + TOPSEG_ELEMS;            // 262144 floats

    // workspace layout: Q (64x512), K (512x512), index (128 ints)  ~= 1.15 MB
    float* Qbuf = (float*)d_ws;
    float* Kbuf = Qbuf + NB * D512;
    int*   idx  = (int*)(Kbuf + (size_t)NB * NS * D512);

    // 1) seg_fea (also an output)
    segfea_kernel<<<SEGFEA_ELEMS / 256, 256, 0, stream>>>(visual, seg_fea);

    // 2) Q = query @ Wq^T + bq  : M=64  -> 4*32 = 128 tiles, 8 waves/block -> 16 blocks
    gemm_wmma_kernel<<<16, 256, 0, stream>>>(query, Wq, bq, Qbuf, NB);

    // 3) K = seg_fea @ Ws^T + bs : M=512 -> 32*32 = 1024 tiles -> 128 blocks
    gemm_wmma_kernel<<<128, 256, 0, stream>>>(seg_fea, Ws, bs, Kbuf, NB * NS);

    // 4) scores + gumbel argmax -> indices
    score_kernel<<<1, 512, 0, stream>>>(Qbuf, Kbuf, y_gumbel, idx);

    // 5) gather-copy 64 MB via 128-bit accesses (the bandwidth-bound bulk of the task)
    gather_kernel<<<TOPSEG_ELEMS / 4 / 256, 256, 0, stream>>>(
        (const float4*)visual, idx, (float4*)top_seg);
}